// Model_61976378081953
// MI455X (gfx1250) — compile-verified
//
#include <hip/hip_runtime.h>
#include <hip/hip_bf16.h>
#include <math.h>

// ---------------------------------------------------------------------------
// MI455X / gfx1250, wave32. All matrix math via V_WMMA_F32_16X16X4_F32:
//   D[16x16] = A[16x4] * B[4x16] + C   (fp32, exact match of fp32 reference)
// A layout : lanes 0-15 M=lane, lanes16-31 same M; VGPR0={K0|K2}, VGPR1={K1|K3}
// B layout : lanes 0-15 N=lane, lanes16-31 same N; VGPR0={K0|K2}, VGPR1={K1|K3}
// C/D      : VGPR r -> (M=r, N=lane&15) lanes0-15 ; (M=8+r) lanes16-31
// ---------------------------------------------------------------------------

typedef float v2f __attribute__((ext_vector_type(2)));
typedef float v8f __attribute__((ext_vector_type(8)));

__device__ __forceinline__ v8f wmma_f32x4(v2f a, v2f b, v8f c) {
  return __builtin_amdgcn_wmma_f32_16x16x4_f32(false, a, false, b, (short)0, c,
                                               false, false);
}

#define NB    64     // batch (sequence length of SA)
#define DSA   1280
#define DCA   8
#define GP    8
#define EPSLN 1e-5f

// ---------------------------------------------------------------------------
// Generic GEMM: out[M,N] = op( A[M,K] @ W[N,K]^T + bias[N] (+ res[M,N]) )
// One wave per 16x16 output tile, K consumed 4 at a time by WMMA.
// ---------------------------------------------------------------------------
template <bool RELU, bool RES>
__global__ __launch_bounds__(32) void gemm16_wmma(
    const float* __restrict__ A, const float* __restrict__ W,
    const float* __restrict__ bias, const float* __restrict__ res,
    float* __restrict__ out, int M, int N, int K) {
  const int lane = threadIdx.x & 31;
  const int l15 = lane & 15, hi = lane >> 4;
  const int n0 = blockIdx.x * 16;
  const int m0 = blockIdx.y * 16;
  const int col = n0 + l15;
  const bool bvalid = (col < N);

  const float* arow = A + (size_t)(m0 + l15) * K;
  const float* wrow = W + (size_t)(bvalid ? col : 0) * K;

  v8f c = {};
  for (int k = 0; k < K; k += 4) {
    v2f a, b;
    a.x = arow[k + 2 * hi];
    a.y = arow[k + 2 * hi + 1];
    if (bvalid) {
      b.x = wrow[k + 2 * hi];
      b.y = wrow[k + 2 * hi + 1];
    } else {
      b.x = 0.f; b.y = 0.f;
    }
    c = wmma_f32x4(a, b, c);
  }
  if (bvalid) {
    float bb = bias[col];
#pragma unroll
    for (int r = 0; r < 8; ++r) {
      int row = m0 + r + 8 * hi;
      float v = c[r] + bb;
      if (RES) v += res[(size_t)row * N + col];
      if (RELU) v = fmaxf(v, 0.f);
      out[(size_t)row * N + col] = v;
    }
  }
}

// ---------------------------------------------------------------------------
// SA attention: 8 heads, S=64, dh=160. One block per head, 4 waves.
// Scores (64x64) resident in LDS; WMMA for QK^T and P@V.
// ---------------------------------------------------------------------------
__global__ __launch_bounds__(128) void sa_attn_kernel(
    const float* __restrict__ qkv, float* __restrict__ attn_out) {
  const int h = blockIdx.x;
  const int wave = threadIdx.x >> 5;
  const int lane = threadIdx.x & 31;
  const int l15 = lane & 15, hi = lane >> 4;
  __shared__ float sc[64 * 64];

  const float* Q = qkv + h * 160;
  const float* Kh = qkv + 1280 + h * 160;
  const float* V = qkv + 2560 + h * 160;
  const int m0 = wave * 16;
  const float scale = 0.07905694150420949f;  // 160^-0.5

  for (int nt = 0; nt < 4; ++nt) {
    const int n0 = nt * 16;
    v8f c = {};
    for (int k = 0; k < 160; k += 4) {
      v2f a, b;
      a.x = Q[(size_t)(m0 + l15) * 3840 + k + 2 * hi];
      a.y = Q[(size_t)(m0 + l15) * 3840 + k + 2 * hi + 1];
      b.x = Kh[(size_t)(n0 + l15) * 3840 + k + 2 * hi];
      b.y = Kh[(size_t)(n0 + l15) * 3840 + k + 2 * hi + 1];
      c = wmma_f32x4(a, b, c);
    }
#pragma unroll
    for (int r = 0; r < 8; ++r)
      sc[(m0 + r + 8 * hi) * 64 + n0 + l15] = c[r] * scale;
  }
  __syncthreads();

  // softmax: one thread per row
  if (threadIdx.x < 64) {
    float* row = sc + threadIdx.x * 64;
    float mx = -1e30f;
    for (int j = 0; j < 64; ++j) mx = fmaxf(mx, row[j]);
    float s = 0.f;
    for (int j = 0; j < 64; ++j) { float e = expf(row[j] - mx); row[j] = e; s += e; }
    float inv = 1.f / s;
    for (int j = 0; j < 64; ++j) row[j] *= inv;
  }
  __syncthreads();

  // O = P @ V : N=160 -> 10 tiles, K=64
  for (int nt = 0; nt < 10; ++nt) {
    const int n0 = nt * 16;
    v8f c = {};
    for (int k = 0; k < 64; k += 4) {
      v2f a, b;
      a.x = sc[(m0 + l15) * 64 + k + 2 * hi];
      a.y = sc[(m0 + l15) * 64 + k + 2 * hi + 1];
      b.x = V[(size_t)(k + 2 * hi) * 3840 + n0 + l15];
      b.y = V[(size_t)(k + 2 * hi + 1) * 3840 + n0 + l15];
      c = wmma_f32x4(a, b, c);
    }
#pragma unroll
    for (int r = 0; r < 8; ++r)
      attn_out[(size_t)(m0 + r + 8 * hi) * 1280 + h * 160 + n0 + l15] = c[r];
  }
}

// ---------------------------------------------------------------------------
// Row LayerNorm over N=1280
// ---------------------------------------------------------------------------
__global__ __launch_bounds__(256) void ln_row_kernel(
    const float* __restrict__ in, const float* __restrict__ g,
    const float* __restrict__ bta, float* __restrict__ out, int N) {
  const int row = blockIdx.x, tid = threadIdx.x;
  const float* x = in + (size_t)row * N;
  __shared__ float red[256];

  float s = 0.f;
  for (int j = tid; j < N; j += 256) s += x[j];
  red[tid] = s; __syncthreads();
  for (int st = 128; st > 0; st >>= 1) {
    if (tid < st) red[tid] += red[tid + st];
    __syncthreads();
  }
  const float mean = red[0] / N;
  __syncthreads();

  float v = 0.f;
  for (int j = tid; j < N; j += 256) { float d = x[j] - mean; v += d * d; }
  red[tid] = v; __syncthreads();
  for (int st = 128; st > 0; st >>= 1) {
    if (tid < st) red[tid] += red[tid + st];
    __syncthreads();
  }
  const float inv = rsqrtf(red[0] / N + EPSLN);
  for (int j = tid; j < N; j += 256)
    out[(size_t)row * N + j] = (x[j] - mean) * inv * g[j] + bta[j];
}

// ---------------------------------------------------------------------------
// Group gather + raw reshape: grouped[b][s][d] = gathered[b].flat[s*8+d]
// ---------------------------------------------------------------------------
__global__ __launch_bounds__(256) void gather_kernel(
    const float* __restrict__ x, const int* __restrict__ pid,
    float* __restrict__ grouped) {
  const int b = blockIdx.x;
  const int myid = pid[b];
  int idx[8]; int cnt = 0;
  for (int j = 0; j < 64; ++j)
    if (pid[j] == myid && cnt < 8) idx[cnt++] = j;
  for (int f = threadIdx.x; f < 8 * 1280; f += 256) {
    int r = f / 1280, c = f - r * 1280;
    grouped[(size_t)b * 10240 + f] =
        (r < cnt) ? x[(size_t)idx[r] * 1280 + c] : 0.f;
  }
}

// ---------------------------------------------------------------------------
// CA attention: per (qtile, head, batch). S=1280, dh=4 -> QK^T tile = ONE
// 16x16x4 WMMA. 16x1280 score strip in (dynamic) LDS; two-pass softmax;
// P@V accumulated with chained WMMAs.
// ---------------------------------------------------------------------------
__global__ __launch_bounds__(32) void ca_attn_kernel(
    const float* __restrict__ qkvca, float* __restrict__ out) {
  extern __shared__ float sc[];  // 16 * 1280 floats = 80 KB
  const int qt = blockIdx.x, h = blockIdx.y, b = blockIdx.z;
  const int lane = threadIdx.x;
  const int l15 = lane & 15, hi = lane >> 4;
  const float* base = qkvca + (size_t)b * 1280 * 24;
  const int q0 = qt * 16;

  v2f aq;  // Q tile, K=4 total: loaded once
  aq.x = base[(size_t)(q0 + l15) * 24 + h * 4 + 2 * hi];
  aq.y = base[(size_t)(q0 + l15) * 24 + h * 4 + 2 * hi + 1];

  for (int kt = 0; kt < 80; ++kt) {
    v2f bm;
    const int krow = kt * 16 + l15;
    bm.x = base[(size_t)krow * 24 + 8 + h * 4 + 2 * hi];
    bm.y = base[(size_t)krow * 24 + 8 + h * 4 + 2 * hi + 1];
    v8f c = {};
    c = wmma_f32x4(aq, bm, c);
#pragma unroll
    for (int r = 0; r < 8; ++r)
      sc[(r + 8 * hi) * 1280 + kt * 16 + l15] = c[r] * 0.5f;  // dh=4 scale
  }
  __syncthreads();

  // softmax over 1280: row = l15, each half-wave handles 640 cols
  {
    float* row = sc + l15 * 1280 + hi * 640;
    float mx = -1e30f;
    for (int j = 0; j < 640; ++j) mx = fmaxf(mx, row[j]);
    mx = fmaxf(mx, __shfl_xor(mx, 16, 32));
    float s = 0.f;
    for (int j = 0; j < 640; ++j) { float e = expf(row[j] - mx); row[j] = e; s += e; }
    s += __shfl_xor(s, 16, 32);
    float inv = 1.f / s;
    for (int j = 0; j < 640; ++j) row[j] *= inv;
  }
  __syncthreads();

  // O = P[16,1280] @ V[1280,4]  (V padded to 16 cols with zeros)
  v8f o = {};
  const int col = l15;
  for (int kt = 0; kt < 80; ++kt) {
#pragma unroll
    for (int ks = 0; ks < 4; ++ks) {
      const int k = kt * 16 + ks * 4;
      v2f a, bm;
      a.x = sc[l15 * 1280 + k + 2 * hi];
      a.y = sc[l15 * 1280 + k + 2 * hi + 1];
      if (col < 4) {
        bm.x = base[(size_t)(k + 2 * hi) * 24 + 16 + h * 4 + col];
        bm.y = base[(size_t)(k + 2 * hi + 1) * 24 + 16 + h * 4 + col];
      } else {
        bm.x = 0.f; bm.y = 0.f;
      }
      o = wmma_f32x4(a, bm, o);
    }
  }
  if (col < 4) {
#pragma unroll
    for (int r = 0; r < 8; ++r)
      out[((size_t)b * 1280 + q0 + r + 8 * hi) * 8 + h * 4 + col] = o[r];
  }
}

// ---------------------------------------------------------------------------
// CA out-proj (8x8) + residual(grouped) + LayerNorm(8). One thread per row.
// ---------------------------------------------------------------------------
__global__ __launch_bounds__(256) void ca_proj_ln_kernel(
    const float* __restrict__ attn, const float* __restrict__ grouped,
    const float* __restrict__ ow, const float* __restrict__ ob,
    const float* __restrict__ g1, const float* __restrict__ b1,
    float* __restrict__ out) {
  const size_t row = (size_t)blockIdx.x * blockDim.x + threadIdx.x;
  if (row >= (size_t)NB * 1280) return;
  float a[8], o[8];
#pragma unroll
  for (int e = 0; e < 8; ++e) a[e] = attn[row * 8 + e];
  float mean = 0.f;
#pragma unroll
  for (int d = 0; d < 8; ++d) {
    float t = ob[d] + grouped[row * 8 + d];
#pragma unroll
    for (int e = 0; e < 8; ++e) t += a[e] * ow[d * 8 + e];
    o[d] = t; mean += t;
  }
  mean *= 0.125f;
  float var = 0.f;
#pragma unroll
  for (int d = 0; d < 8; ++d) { float df = o[d] - mean; var += df * df; }
  var *= 0.125f;
  const float inv = rsqrtf(var + EPSLN);
#pragma unroll
  for (int d = 0; d < 8; ++d)
    out[row * 8 + d] = (o[d] - mean) * inv * g1[d] + b1[d];
}

// ---------------------------------------------------------------------------
// CA FFN fused: h = relu(x1@l1w^T + b)[16x1280 tile-wise], out = h@l2w^T + b
//               + residual, + LayerNorm(8). Hidden never hits HBM: each 16x16
//               hidden tile is relu'd, bounced through 1KB LDS (C->A layout),
//               and consumed by 4 chained WMMAs.
// ---------------------------------------------------------------------------
__global__ __launch_bounds__(32) void ca_ffn_kernel(
    const float* __restrict__ x1, const float* __restrict__ l1w,
    const float* __restrict__ l1b, const float* __restrict__ l2w,
    const float* __restrict__ l2b, const float* __restrict__ g2,
    const float* __restrict__ b2, float* __restrict__ out) {
  __shared__ float P[16 * 16];
  const int lane = threadIdx.x;
  const int l15 = lane & 15, hi = lane >> 4;
  const size_t rowbase = (size_t)blockIdx.y * 1280 + blockIdx.x * 16;

  v2f a0, a1;  // x1 tile [16 x 8] as two K=4 A operands
  a0.x = x1[(rowbase + l15) * 8 + 2 * hi];
  a0.y = x1[(rowbase + l15) * 8 + 2 * hi + 1];
  a1.x = x1[(rowbase + l15) * 8 + 4 + 2 * hi];
  a1.y = x1[(rowbase + l15) * 8 + 4 + 2 * hi + 1];

  v8f acc = {};
  for (int ft = 0; ft < 80; ++ft) {
    const int f0 = ft * 16;
    v2f b0, b1v;
    b0.x = l1w[(size_t)(f0 + l15) * 8 + 2 * hi];
    b0.y = l1w[(size_t)(f0 + l15) * 8 + 2 * hi + 1];
    b1v.x = l1w[(size_t)(f0 + l15) * 8 + 4 + 2 * hi];
    b1v.y = l1w[(size_t)(f0 + l15) * 8 + 4 + 2 * hi + 1];
    v8f hreg = {};
    hreg = wmma_f32x4(a0, b0, hreg);
    hreg = wmma_f32x4(a1, b1v, hreg);
    const float bias = l1b[f0 + l15];
#pragma unroll
    for (int r = 0; r < 8; ++r)
      P[(r + 8 * hi) * 16 + l15] = fmaxf(hreg[r] + bias, 0.f);
    __syncthreads();
#pragma unroll
    for (int ks = 0; ks < 4; ++ks) {
      const int k = ks * 4 + 2 * hi;
      v2f a, bm;
      a.x = P[l15 * 16 + k];
      a.y = P[l15 * 16 + k + 1];
      const int d = l15;
      if (d < 8) {
        bm.x = l2w[(size_t)d * 1280 + f0 + k];
        bm.y = l2w[(size_t)d * 1280 + f0 + k + 1];
      } else {
        bm.x = 0.f; bm.y = 0.f;
      }
      acc = wmma_f32x4(a, bm, acc);
    }
    __syncthreads();
  }

  // epilogue: bias + residual + LayerNorm over d=0..7 (8-lane groups)
  const int d = l15;
  const int dc = d < 8 ? d : 0;
#pragma unroll
  for (int r = 0; r < 8; ++r) {
    const size_t row = rowbase + r + 8 * hi;
    float v = (d < 8) ? (acc[r] + l2b[dc] + x1[row * 8 + dc]) : 0.f;
    float s = v;
    s += __shfl_xor(s, 1, 8); s += __shfl_xor(s, 2, 8); s += __shfl_xor(s, 4, 8);
    const float mean = s * 0.125f;
    float df = v - mean;
    float q = df * df;
    q += __shfl_xor(q, 1, 8); q += __shfl_xor(q, 2, 8); q += __shfl_xor(q, 4, 8);
    const float y = df * rsqrtf(q * 0.125f + EPSLN) * g2[dc] + b2[dc];
    if (d < 8) out[row * 8 + d] = y;
  }
}

// ---------------------------------------------------------------------------
// Pool(sum over 8) + concat aux + two linear heads.
// ---------------------------------------------------------------------------
__global__ __launch_bounds__(128) void head_kernel(
    const float* __restrict__ cax2, const float* __restrict__ aux,
    const float* __restrict__ clfw, const float* __restrict__ clfb,
    const float* __restrict__ auxw, const float* __restrict__ auxb,
    float* __restrict__ out) {
  const int b = blockIdx.x, tid = threadIdx.x;
  __shared__ float r1[128], r2[128];
  float a1 = 0.f, a2 = 0.f;
  for (int i = tid; i < 1280; i += 128) {
    const float* p = cax2 + ((size_t)b * 1280 + i) * 8;
    float s = p[0] + p[1] + p[2] + p[3] + p[4] + p[5] + p[6] + p[7];
    a1 += s * clfw[i];
    a2 += s * auxw[i];
  }
  r1[tid] = a1; r2[tid] = a2; __syncthreads();
  for (int st = 64; st > 0; st >>= 1) {
    if (tid < st) { r1[tid] += r1[tid + st]; r2[tid] += r2[tid + st]; }
    __syncthreads();
  }
  if (tid == 0) {
    const float x0 = aux[b], x1 = aux[64 + b];
    const float cancer = r1[0] + clfw[1280] * x0 + clfw[1281] * x1 + clfb[0];
    const float inv = r2[0] + auxw[1280] * x0 + auxw[1281] * x1 +
                      auxw[1282] * cancer + auxb[0];
    out[b] = cancer;
    out[64 + b] = inv;
  }
}

// ---------------------------------------------------------------------------
extern "C" void kernel_launch(void* const* d_in, const int* in_sizes, int n_in,
                              void* d_out, int out_size, void* d_ws,
                              size_t ws_size, hipStream_t stream) {
  (void)in_sizes; (void)n_in; (void)out_size; (void)ws_size;
  const float* x        = (const float*)d_in[0];
  const float* aux      = (const float*)d_in[1];
  const int*   pid      = (const int*)d_in[2];
  // sa params (insertion order)
  const float* sa_inw   = (const float*)d_in[3];
  const float* sa_inb   = (const float*)d_in[4];
  const float* sa_ow    = (const float*)d_in[5];
  const float* sa_obv   = (const float*)d_in[6];
  const float* sa_l1w   = (const float*)d_in[7];
  const float* sa_l1b   = (const float*)d_in[8];
  const float* sa_l2w   = (const float*)d_in[9];
  const float* sa_l2b   = (const float*)d_in[10];
  const float* sa_ln1g  = (const float*)d_in[11];
  const float* sa_ln1b  = (const float*)d_in[12];
  const float* sa_ln2g  = (const float*)d_in[13];
  const float* sa_ln2b  = (const float*)d_in[14];
  // ca params
  const float* ca_inw   = (const float*)d_in[15];
  const float* ca_inb   = (const float*)d_in[16];
  const float* ca_ow    = (const float*)d_in[17];
  const float* ca_obv   = (const float*)d_in[18];
  const float* ca_l1w   = (const float*)d_in[19];
  const float* ca_l1b   = (const float*)d_in[20];
  const float* ca_l2w   = (const float*)d_in[21];
  const float* ca_l2b   = (const float*)d_in[22];
  const float* ca_ln1g  = (const float*)d_in[23];
  const float* ca_ln1b  = (const float*)d_in[24];
  const float* ca_ln2g  = (const float*)d_in[25];
  const float* ca_ln2b  = (const float*)d_in[26];
  const float* clfw     = (const float*)d_in[27];
  const float* clfb     = (const float*)d_in[28];
  const float* auxw     = (const float*)d_in[29];
  const float* auxb     = (const float*)d_in[30];
  float* out = (float*)d_out;

  float* ws = (float*)d_ws;
  float* qkv_sa  = ws;                     // 64*3840
  float* attn_sa = qkv_sa  + 245760;       // 64*1280
  float* tmp     = attn_sa + 81920;        // 64*1280 (pre-LN scratch)
  float* x1sa    = tmp     + 81920;
  float* x2sa    = x1sa    + 81920;
  float* hff     = x2sa    + 81920;        // FFN hidden
  float* grouped = hff     + 81920;        // 64*1280*8
  float* qkvca   = grouped + 655360;       // 81920*24
  float* ca_attn = qkvca   + 1966080;      // 81920*8
  float* ca_x1   = ca_attn + 655360;       // 81920*8
  float* ca_x2   = ca_x1   + 655360;       // 81920*8

  // ---- SA encoder layer ----
  gemm16_wmma<false, false><<<dim3(240, 4), 32, 0, stream>>>(
      x, sa_inw, sa_inb, nullptr, qkv_sa, 64, 3840, 1280);
  sa_attn_kernel<<<8, 128, 0, stream>>>(qkv_sa, attn_sa);
  gemm16_wmma<false, true><<<dim3(80, 4), 32, 0, stream>>>(
      attn_sa, sa_ow, sa_obv, x, tmp, 64, 1280, 1280);
  ln_row_kernel<<<64, 256, 0, stream>>>(tmp, sa_ln1g, sa_ln1b, x1sa, 1280);
  gemm16_wmma<true, false><<<dim3(80, 4), 32, 0, stream>>>(
      x1sa, sa_l1w, sa_l1b, nullptr, hff, 64, 1280, 1280);
  gemm16_wmma<false, true><<<dim3(80, 4), 32, 0, stream>>>(
      hff, sa_l2w, sa_l2b, x1sa, tmp, 64, 1280, 1280);
  ln_row_kernel<<<64, 256, 0, stream>>>(tmp, sa_ln2g, sa_ln2b, x2sa, 1280);

  // ---- group gather + raw reshape ----
  gather_kernel<<<64, 256, 0, stream>>>(x2sa, pid, grouped);

  // ---- CA encoder layer (vmapped over 64 batches) ----
  gemm16_wmma<false, false><<<dim3(2, 5120), 32, 0, stream>>>(
      grouped, ca_inw, ca_inb, nullptr, qkvca, 64 * 1280, 24, 8);
  ca_attn_kernel<<<dim3(80, 2, 64), 32, 16 * 1280 * sizeof(float), stream>>>(
      qkvca, ca_attn);
  ca_proj_ln_kernel<<<320, 256, 0, stream>>>(
      ca_attn, grouped, ca_ow, ca_obv, ca_ln1g, ca_ln1b, ca_x1);
  ca_ffn_kernel<<<dim3(80, 64), 32, 0, stream>>>(
      ca_x1, ca_l1w, ca_l1b, ca_l2w, ca_l2b, ca_ln2g, ca_ln2b, ca_x2);

  // ---- heads ----
  head_kernel<<<64, 128, 0, stream>>>(ca_x2, aux, clfw, clfb, auxw, auxb, out);
}